// Coupling_23467701306036
// MI455X (gfx1250) — compile-verified
//
#include <hip/hip_runtime.h>

// ---------------------------------------------------------------------------
// Flow++ mixture-of-logistics coupling (forward), MI455X / gfx1250.
// B=16, K=16, C=64, L=2048.  ~416 MiB streamed -> HBM-bound (~18us floor).
// fp32 streaming with non-temporal hints; K-reduction in registers;
// per-batch sldj via deterministic LDS tree + V_WMMA_F32_16X16X4_F32 reduce.
// ---------------------------------------------------------------------------

typedef float v2f __attribute__((ext_vector_type(2)));
typedef float v4f __attribute__((ext_vector_type(4)));
typedef float v8f __attribute__((ext_vector_type(8)));

#define KMIX 16
#define NBATCH 16
#define INNER 131072                      // C*L = 64*2048
#define NPOINTS (NBATCH * INNER)          // 2,097,152
#define BLOCKT 256
#define PPT 4                             // points per thread (float4)
#define PTS_PER_BLOCK (BLOCKT * PPT)      // 1024
#define BLOCKS_PER_BATCH (INNER / PTS_PER_BLOCK)  // 128
#define NGRID (NBATCH * BLOCKS_PER_BATCH)         // 2048
#define STRIDE4 (INNER / 4)               // 32768 (v4f stride between k planes)

__device__ __forceinline__ float fexp(float x) { return __expf(x); }
__device__ __forceinline__ float flog(float x) { return __logf(x); }
__device__ __forceinline__ float frcp(float x) { return __builtin_amdgcn_rcpf(x); }

// ---------------------------------------------------------------------------
// Main streaming kernel: one thread = 4 consecutive L-points.
// ---------------------------------------------------------------------------
__global__ __launch_bounds__(BLOCKT) void mol_coupling_kernel(
    const float* __restrict__ x, const float* __restrict__ aa,
    const float* __restrict__ bb, const float* __restrict__ pi,
    const float* __restrict__ mu, const float* __restrict__ s,
    float* __restrict__ out, float* __restrict__ ws)
{
    const int tid = threadIdx.x;
    const int b   = blockIdx.x >> 7;          // / BLOCKS_PER_BATCH
    const int lb  = blockIdx.x & 127;
    const int p0  = lb * PTS_PER_BLOCK + tid * PPT;     // inner offset (mult of 4)
    const int xi  = b * (INNER / 4) + (p0 >> 2);        // v4f idx into x/a/b/out
    const int gi  = b * (KMIX * INNER / 4) + (p0 >> 2); // v4f idx into pi/mu/s (k=0)

    const v4f* x4  = (const v4f*)x;
    const v4f* a4  = (const v4f*)aa;
    const v4f* b4  = (const v4f*)bb;
    const v4f* pi4 = (const v4f*)pi;
    const v4f* mu4 = (const v4f*)mu;
    const v4f* s4  = (const v4f*)s;

    // Non-temporal: every byte is used exactly once; don't thrash 192MB L2.
    v4f xv = __builtin_nontemporal_load(x4 + xi);
    v4f av = __builtin_nontemporal_load(a4 + xi);
    v4f bv = __builtin_nontemporal_load(b4 + xi);

    // ---- pass 1: pi -> unnormalized softmax weights e_k = exp(pi_k - max) ----
    v4f e[KMIX];
#pragma unroll
    for (int k = 0; k < KMIX; ++k)
        e[k] = __builtin_nontemporal_load(pi4 + gi + k * STRIDE4);

    float sume[PPT];
#pragma unroll
    for (int p = 0; p < PPT; ++p) {
        float m = e[0][p];
#pragma unroll
        for (int k = 1; k < KMIX; ++k) m = fmaxf(m, e[k][p]);
        float se = 0.0f;
#pragma unroll
        for (int k = 0; k < KMIX; ++k) {
            float t = fexp(e[k][p] - m);
            e[k][p] = t;            // reuse register storage as e_k
            se += t;
        }
        sume[p] = se;
    }

    // ---- pass 2: mixture CDF (u) and mixture PDF accumulators ----
    // per term: w = exp(-|z|), r = 1/(1+w)
    //   sigmoid(z)              = z>=0 ? r : w*r
    //   exp(z - s - 2*softplus) = w * r^2 * exp(-s)
    float uacc[PPT] = {0.f, 0.f, 0.f, 0.f};
    float pacc[PPT] = {0.f, 0.f, 0.f, 0.f};
#pragma unroll
    for (int k = 0; k < KMIX; ++k) {
        v4f muv = __builtin_nontemporal_load(mu4 + gi + k * STRIDE4);
        v4f sv  = __builtin_nontemporal_load(s4  + gi + k * STRIDE4);
#pragma unroll
        for (int p = 0; p < PPT; ++p) {
            float es  = fexp(-sv[p]);
            float z   = (xv[p] - muv[p]) * es;
            float w   = fexp(-fabsf(z));
            float r   = frcp(1.0f + w);
            float sig = (z >= 0.0f) ? r : w * r;
            float pdf = w * r * r * es;
            uacc[p] += e[k][p] * sig;
            pacc[p] += e[k][p] * pdf;
        }
    }

    // ---- epilogue: logit transform + log-dets ----
    v4f ov;
    float ldj = 0.0f;
#pragma unroll
    for (int p = 0; p < PPT; ++p) {
        float rn = frcp(sume[p]);
        float u    = uacc[p] * rn;           // exp(mixture_log_cdf) in (0,1)
        float sum2 = pacc[p] * rn;           // exp(mixture_log_pdf)
        float logistic_ldj = flog(sum2);
        float v = frcp(u) - 1.0f;            // (1-u)/u
        float out_pre   = -flog(fmaxf(v, 1e-22f));
        float scale_ldj = -flog(fmaxf(u, 1e-22f)) - flog(fmaxf(1.0f - u, 1e-22f));
        // scale = sigmoid(a + 2) + (1 - sigmoid(2))
        float sc = frcp(1.0f + fexp(-(av[p] + 2.0f))) + 0.1192029220221177f;
        ldj += logistic_ldj + scale_ldj + flog(sc);
        ov[p] = (out_pre + bv[p]) * sc;
    }
    __builtin_nontemporal_store(ov, (v4f*)out + xi);

    // ---- deterministic block reduction of ldj -> ws[blockIdx.x] ----
    __shared__ float red[BLOCKT];
    red[tid] = ldj;
    __syncthreads();
#pragma unroll
    for (int off = BLOCKT / 2; off > 0; off >>= 1) {
        if (tid < off) red[tid] += red[tid + off];
        __syncthreads();
    }
    if (tid == 0) ws[blockIdx.x] = red[0];
}

// ---------------------------------------------------------------------------
// Per-batch reduction of the 128 block partials using V_WMMA_F32_16X16X4_F32.
// B = all-ones  =>  D[:,j] = row-sums of A (+C);  two chained WMMAs sum all
// 128 values; column 0 lives on lanes 0 (M=0..7) and 16 (M=8..15).
// One wave per batch; EXEC is all ones (no divergence before the WMMAs).
// ---------------------------------------------------------------------------
__global__ __launch_bounds__(32) void sldj_wmma_reduce(
    const float* __restrict__ ws, const float* __restrict__ sldj_in,
    float* __restrict__ out_sldj)
{
    const int b    = blockIdx.x;
    const int lane = threadIdx.x;
    const float* p = ws + b * BLOCKS_PER_BATCH;

    v2f a1, a2, ones;
    a1[0] = p[lane];        a1[1] = p[lane + 32];
    a2[0] = p[lane + 64];   a2[1] = p[lane + 96];
    ones[0] = 1.0f;         ones[1] = 1.0f;

    v8f c = {0.f, 0.f, 0.f, 0.f, 0.f, 0.f, 0.f, 0.f};
    c = __builtin_amdgcn_wmma_f32_16x16x4_f32(false, a1, false, ones,
                                              (short)0, c, false, false);
    c = __builtin_amdgcn_wmma_f32_16x16x4_f32(false, a2, false, ones,
                                              (short)0, c, false, false);

    // column 0 of D: 8 VGPRs on lane 0 (M=0..7) + 8 VGPRs on lane 16 (M=8..15)
    float s8 = c[0] + c[1] + c[2] + c[3] + c[4] + c[5] + c[6] + c[7];
    float other = __shfl(s8, lane ^ 16, 32);
    if (lane == 0) out_sldj[b] = s8 + other + sldj_in[b];
}

// ---------------------------------------------------------------------------
extern "C" void kernel_launch(void* const* d_in, const int* in_sizes, int n_in,
                              void* d_out, int out_size, void* d_ws, size_t ws_size,
                              hipStream_t stream)
{
    const float* x    = (const float*)d_in[0];
    const float* aa   = (const float*)d_in[1];
    const float* bb   = (const float*)d_in[2];
    const float* pi   = (const float*)d_in[3];
    const float* mu   = (const float*)d_in[4];
    const float* s    = (const float*)d_in[5];
    const float* sldj = (const float*)d_in[6];

    float* out = (float*)d_out;              // [NPOINTS] out, then [NBATCH] sldj
    float* ws  = (float*)d_ws;               // NGRID block partials (8 KiB)

    mol_coupling_kernel<<<NGRID, BLOCKT, 0, stream>>>(x, aa, bb, pi, mu, s, out, ws);
    sldj_wmma_reduce<<<NBATCH, 32, 0, stream>>>(ws, sldj, out + NPOINTS);
}